// SingleLayerSNN_52845277610450
// MI455X (gfx1250) — compile-verified
//
#include <hip/hip_runtime.h>

// SNN forward, fused GEMM + neuron scan for MI455X (gfx1250).
// Grid: 128 WGs (8 M-tiles x 16 N-blocks of 32 cols), 128 threads (4 waves).
// Waves: (wn, kh) = (N sub-tile 0..1, K half 0..1). Each 16x16 output tile is
// computed by two waves (K split 512+512); kh=1 ships partials through LDS,
// kh=0 reduces + owns neuron state. W slab [1024 x 32] f32 persists in LDS.

typedef float v2f __attribute__((ext_vector_type(2)));
typedef float v8f __attribute__((ext_vector_type(8)));

static constexpr int kF = 1024;   // IN_FEATURES
static constexpr int kC = 512;    // OUT_CLASSES
static constexpr int kT = 256;    // T_STEPS

static constexpr float V_TH      = 0.4f;
static constexpr float LEAK      = 0.01f;
static constexpr float REFR_LEAK = 0.01f;
static constexpr float REFRAC    = 2.0f;

static constexpr int kSlabFloats = 1024 * 32;            // 128 KB W slab
static constexpr int kPartFloats = 2 * 2 * 256;          // 2 bufs x 2 tiles x 256
static constexpr int kLdsBytes   = (kSlabFloats + kPartFloats) * 4;

__global__ __launch_bounds__(128)
void snn_fused_wmma(const float* __restrict__ x,   // [B, T, F]
                    const float* __restrict__ W,   // [F, C]
                    float* __restrict__ out) {     // [B, C]
  extern __shared__ float Wl[];          // slab: [(k>>1)][n*2 + (k&1)], 64 floats/row
  float* pbuf = Wl + kSlabFloats;        // partial-tile exchange buffers

  const int tid  = threadIdx.x;
  const int lane = tid & 31;
  const int wv   = tid >> 5;             // 0..3
  const int wn   = wv & 1;               // N sub-tile within WG
  const int kh   = wv >> 1;              // K half (0 owns state)
  const int bm   = blockIdx.x >> 4;      // 0..7
  const int bn   = blockIdx.x & 15;      // 0..15
  const int b0   = bm << 4;
  const int n0   = bn << 5;

  // ---- One-time W slab fill: async global->LDS DMA (ASYNCcnt path).
  // Wl[(kk>>1)*64 + n*2 + (kk&1)] = W[kk, n0+n]
  for (int i = 0; i < 256; ++i) {
    int idx = tid + (i << 7);
    int kk  = idx >> 5;
    int n   = idx & 31;
    unsigned ldsOff =
        (unsigned)(size_t)(Wl + ((kk >> 1) << 6) + (n << 1) + (kk & 1));
    unsigned long long g =
        (unsigned long long)(size_t)(W + (size_t)kk * kC + n0 + n);
    asm volatile("global_load_async_to_lds_b32 %0, %1, off"
                 :: "v"(ldsOff), "v"(g) : "memory");
  }
  asm volatile("s_wait_asynccnt 0x0" ::: "memory");
  __syncthreads();

  // ---- WMMA f32 16x16x4 fragment addressing
  // A: lane row m = lane&15; lanes 0-15 K={0,1}, 16-31 K={2,3}
  const int m     = lane & 15;
  const int khalf = (lane >> 4) << 1;                    // 0 or 2
  // B: lane col n = lane&15; pair rows {khalf, khalf+1} of this wave's K half
  const int n_l   = (wn << 4) + (lane & 15);             // 0..31 in slab
  const float* bbase = Wl + (n_l << 1) + ((kh << 8) + (khalf >> 1)) * 64;

  const size_t xrowBase = (size_t)(b0 + m) * kT * kF + (kh << 9) + khalf;

  // ---- Neuron state (C/D layout: VGPR r -> M = r + 8*(lane>=16), N = lane&15)
  float Vv[8], rf[8], sc[8];
  int   md[8];
  #pragma unroll
  for (int r = 0; r < 8; ++r) { Vv[r] = 0.f; rf[r] = 0.f; sc[r] = 0.f; md[r] = 0; }

  for (int t = 0; t < kT; ++t) {
    const float* xa = x + xrowBase + (size_t)t * kF;
    if (t + 1 < kT) __builtin_prefetch(xa + kF, 0, 0);

    v8f acc = {0.f, 0.f, 0.f, 0.f, 0.f, 0.f, 0.f, 0.f};
    #pragma unroll 8
    for (int k = 0; k < 512; k += 4) {
      v2f a   = *(const v2f*)(xa + k);                       // global b64
      v2f bfr = *(const v2f*)(bbase + ((k >> 1) << 6));      // ds b64 (pairs)
      acc = __builtin_amdgcn_wmma_f32_16x16x4_f32(
          false, a, false, bfr, (short)0, acc, false, false);
    }

    // ---- Cross-wave K reduction via LDS (double-buffered, 1 barrier/step)
    float* pb = pbuf + ((((t & 1) << 1) + wn) << 8) + (lane << 3);
    if (kh) *(v8f*)pb = acc;             // partial out (ds_store_b128 x2)
    __syncthreads();

    if (!kh) {
      v8f part = *(const v8f*)pb;
      // ---- Neuron update, exact reference semantics (snapshot masks)
      #pragma unroll
      for (int r = 0; r < 8; ++r) {
        const float I = acc[r] + part[r];
        const bool mstd = (md[r] == 0);
        const bool mabs = (md[r] == 1);
        const bool mrel = (md[r] == 2);
        float v = Vv[r];

        const float vint = fmaxf(v + I - LEAK, 0.0f);
        v = mstd ? vint : v;
        const bool fired = mstd && (v > V_TH);
        sc[r] += fired ? 1.0f : 0.0f;
        v      = fired ? 0.0f : v;
        rf[r]  = fired ? REFRAC : rf[r];

        const float rdec = fmaxf(rf[r] - 1.0f, 0.0f);
        rf[r] = mabs ? rdec : rf[r];
        const bool done = mabs && (rf[r] <= 0.0f);
        v = (mabs && (v < 0.0f)) ? 0.0f : v;

        const float vrel = v + REFR_LEAK;
        v = mrel ? vrel : v;
        v = (mrel && (v < 0.0f)) ? 0.0f : v;
        const bool back = mrel && (v > 0.0f);

        int mdn = md[r];
        mdn = fired ? 1 : mdn;
        mdn = done  ? 2 : mdn;
        mdn = back  ? 0 : mdn;
        md[r] = mdn;
        Vv[r] = v;
      }
    }
  }

  if (!kh) {
    #pragma unroll
    for (int r = 0; r < 8; ++r) {
      const int mr = r + ((lane >> 4) << 3);
      const int nc = n0 + (wn << 4) + (lane & 15);
      out[(size_t)(b0 + mr) * kC + nc] = sc[r];
    }
  }
}

extern "C" void kernel_launch(void* const* d_in, const int* in_sizes, int n_in,
                              void* d_out, int out_size, void* d_ws, size_t ws_size,
                              hipStream_t stream) {
  (void)in_sizes; (void)n_in; (void)d_ws; (void)ws_size; (void)out_size;
  const float* x = (const float*)d_in[0];   // x_seq [128,256,1024] f32
  const float* W = (const float*)d_in[1];   // W [1024,512] f32
  float* out = (float*)d_out;               // [128,512] f32

  dim3 grid(128);    // 8 M-tiles x 16 N-blocks
  dim3 block(128);   // 4 waves: 2 N sub-tiles x 2 K halves
  hipLaunchKernelGGL(snn_fused_wmma, grid, block, kLdsBytes, stream, x, W, out);
}